// IntraModuleAggregation_8435315769784
// MI455X (gfx1250) — compile-verified
//
#include <hip/hip_runtime.h>
#include <math.h>

// ---------------------------------------------------------------------------
// IntraModuleAggregation fused kernel for MI455X (gfx1250, wave32)
//   agg = einsum('btmv,mvw->btmw', x, PA)   -> fp32 WMMA 16x16x4
//   h   = relu(affine(agg))                 -> fused scalar affine
//   out = attention-pool over nodes (tanh MLP + softmax)
// ---------------------------------------------------------------------------

typedef float v2f __attribute__((ext_vector_type(2)));
typedef float v8f __attribute__((ext_vector_type(8)));

#define B_    64
#define T_    512
#define M_    8
#define VM_   50
#define KPAD  52     // K (=v) padded to multiple of 4
#define NPAD  64     // N (=w) padded to 4 tiles of 16
#define TB    128    // T-rows per block (8 waves x 16 rows)

__launch_bounds__(256)
__global__ void intra_module_fused(
    const float* __restrict__ x,        // [B, M*VM, T]
    const float* __restrict__ PA,       // [M, VM, VM]
    const float* __restrict__ conv_w,   // [M]
    const float* __restrict__ conv_b,   // [M]
    const float* __restrict__ bn_gamma, // [M]
    const float* __restrict__ bn_beta,  // [M]
    const float* __restrict__ bn_mean,  // [M]
    const float* __restrict__ bn_var,   // [M]
    const float* __restrict__ w1,       // [M,8]
    const float* __restrict__ b1,       // [M,8]
    const float* __restrict__ w2,       // [M,8]
    const float* __restrict__ b2,       // [M]
    float* __restrict__ out)            // [B, M, T]
{
  // PA transposed: lds_paT[w*KPAD + v] = PA[m][v][w]; zero-padded so that the
  // K=50..51 and w=50..63 pad regions contribute exact zeros to the WMMA.
  __shared__ __align__(16) float lds_paT[NPAD * KPAD];   // 13312 B
  __shared__ __align__(16) float lds_h [TB * VM_];       // 25600 B
  __shared__ __align__(16) float lds_s [TB * VM_];       // 25600 B

  const int tid  = threadIdx.x;
  const int lane = tid & 31;
  const int wv   = tid >> 5;            // wave id 0..7
  const int m    = blockIdx.y;
  const int b    = blockIdx.z;
  const int t0   = blockIdx.x * TB;

  // ---- stage PA_m into LDS (transposed, zero padded) ----
  for (int i = tid; i < NPAD * KPAD; i += 256) lds_paT[i] = 0.0f;
  __syncthreads();
  {
    const float* pam = PA + m * (VM_ * VM_);
    for (int i = tid; i < VM_ * VM_; i += 256) {
      int v = i / VM_, w = i - v * VM_;
      lds_paT[w * KPAD + v] = pam[i];
    }
  }
  __syncthreads();

  // fused conv(1x1) + BatchNorm(eval) affine: h_pre = agg*Aaff + Baff
  const float scale = bn_gamma[m] * rsqrtf(bn_var[m] + 1e-5f);
  const float Aaff  = conv_w[m] * scale;
  const float Baff  = (conv_b[m] - bn_mean[m]) * scale + bn_beta[m];

  // ---- phase 1: per-wave 16x64 fp32 WMMA tile over K=50 (13 steps of 4) ----
  // A(16x4): lane L -> M = L&15, K components = k0 + (L<16 ? 0 : 2) , +1
  // B(4x16): lane L -> N = L&15, same K striping (symmetric layout)
  // C(16x16): VGPR r -> M = r + (L<16 ? 0 : 8), N = L&15
  const int  col    = lane & 15;
  const int  khalf  = (lane < 16) ? 0 : 2;
  const int  trow   = t0 + wv * 16 + col;
  const long xbase  = ((long)b * (M_ * VM_) + (long)m * VM_) * T_ + trow;

  v8f acc[4] = {};   // 4 N-tiles: w = 0..63

#pragma unroll
  for (int k0 = 0; k0 < KPAD; k0 += 4) {
    const int ka = k0 + khalf;
    v2f a;
    a.x = (ka     < VM_) ? x[xbase + (long)ka       * T_] : 0.0f;
    a.y = (ka + 1 < VM_) ? x[xbase + (long)(ka + 1) * T_] : 0.0f;
#pragma unroll
    for (int n = 0; n < 4; ++n) {
      v2f bf = *(const v2f*)&lds_paT[(n * 16 + col) * KPAD + ka];
      acc[n] = __builtin_amdgcn_wmma_f32_16x16x4_f32(
          /*neg_a=*/false, a, /*neg_b=*/false, bf,
          /*c_mod=*/(short)0, acc[n], /*reuse_a=*/false, /*reuse_b=*/false);
    }
  }

  // epilogue: affine + ReLU, scatter valid 16x50 tile to LDS
  const int rowbase = wv * 16 + ((lane < 16) ? 0 : 8);
#pragma unroll
  for (int n = 0; n < 4; ++n) {
    const int w = n * 16 + col;
    if (w < VM_) {
#pragma unroll
      for (int r = 0; r < 8; ++r) {
        float h = fmaxf(fmaf(acc[n][r], Aaff, Baff), 0.0f);
        lds_h[(rowbase + r) * VM_ + w] = h;
      }
    }
  }
  __syncthreads();

  // ---- phase 2: attention pooling, one thread per T-row ----
  if (tid < TB) {
    const int r = tid;
    float w1r[8], b1r[8], w2r[8];
#pragma unroll
    for (int j = 0; j < 8; ++j) {
      w1r[j] = w1[m * 8 + j];
      b1r[j] = b1[m * 8 + j];
      w2r[j] = w2[m * 8 + j];
    }
    const float b2m = b2[m];

    float mx = -3.0e38f;
    for (int w = 0; w < VM_; ++w) {
      const float hv = lds_h[r * VM_ + w];
      float s = b2m;
#pragma unroll
      for (int j = 0; j < 8; ++j)
        s += w2r[j] * tanhf(fmaf(hv, w1r[j], b1r[j]));
      lds_s[r * VM_ + w] = s;
      mx = fmaxf(mx, s);
    }
    float den = 0.0f, num = 0.0f;
    for (int w = 0; w < VM_; ++w) {
      const float e = __expf(lds_s[r * VM_ + w] - mx);
      den += e;
      num += lds_h[r * VM_ + w] * e;
    }
    out[((long)b * M_ + m) * T_ + (t0 + r)] = num / den;
  }
}

extern "C" void kernel_launch(void* const* d_in, const int* in_sizes, int n_in,
                              void* d_out, int out_size, void* d_ws, size_t ws_size,
                              hipStream_t stream) {
  const float* x        = (const float*)d_in[0];
  const float* PA       = (const float*)d_in[1];
  const float* conv_w   = (const float*)d_in[2];
  const float* conv_b   = (const float*)d_in[3];
  const float* bn_gamma = (const float*)d_in[4];
  const float* bn_beta  = (const float*)d_in[5];
  const float* bn_mean  = (const float*)d_in[6];
  const float* bn_var   = (const float*)d_in[7];
  const float* w1       = (const float*)d_in[8];
  const float* b1       = (const float*)d_in[9];
  const float* w2       = (const float*)d_in[10];
  const float* b2       = (const float*)d_in[11];
  float* out = (float*)d_out;

  dim3 grid(T_ / TB, M_, B_);   // (4, 8, 64) blocks
  dim3 block(256);              // 8 wave32's
  intra_module_fused<<<grid, block, 0, stream>>>(
      x, PA, conv_w, conv_b, bn_gamma, bn_beta, bn_mean, bn_var,
      w1, b1, w2, b2, out);
}